// OffsetSubtraction_47785806135946
// MI455X (gfx1250) — compile-verified
//
#include <hip/hip_runtime.h>

// Problem constants (from reference setup_inputs): B=8, W=4096, F=64, d=8.
#define B_  8
#define W_  4096
#define F_  64
#define D_  8                 // max delay; 2*D_+1 = 17 candidates
#define TW  64                // W-rows per block tile
#define HALO D_
#define ROWS (TW + 2*HALO)    // 80 staged rows
#define QPR (F_/4)            // 16 float4 "quads" per row
#define NTHREADS 256

typedef int v4i __attribute__((ext_vector_type(4)));

// ---- gfx1250 async global->LDS path (probe-guarded) ------------------------
#if defined(__has_builtin)
#  if __has_builtin(__builtin_amdgcn_global_load_async_to_lds_b128)
#    define HAVE_ASYNC_LDS 1
#  endif
#endif
#ifndef HAVE_ASYNC_LDS
#  define HAVE_ASYNC_LDS 0
#endif

__device__ __forceinline__ void wait_async_zero() {
#if defined(__has_builtin) && __has_builtin(__builtin_amdgcn_s_wait_asynccnt)
    __builtin_amdgcn_s_wait_asynccnt(0);
#else
    asm volatile("s_wait_asynccnt 0" ::: "memory");
#endif
}

// Update best (signed value with min |value|), strict < keeps the FIRST
// minimal candidate in scan order (matches jnp.argmin tie-breaking).
#define UPD_COMP(c)                                        \
    {                                                      \
        float rr = sv.c - g.c;                             \
        float aa = __builtin_fabsf(rr);                    \
        if (aa < ba.c) { ba.c = aa; bv.c = rr; }           \
    }

#define UPD_ALL() UPD_COMP(x) UPD_COMP(y) UPD_COMP(z) UPD_COMP(w)

__global__ __launch_bounds__(NTHREADS)
void OffsetSubtraction_47785806135946_kernel(const float4* __restrict__ subed,
                                             const float4* __restrict__ sub,
                                             float4* __restrict__ out)
{
    __shared__ float4 lds[ROWS * QPR];   // 80 * 16 * 16B = 20 KB

    const int blk = blockIdx.x;
    const int b   = blk / (W_ / TW);
    const int w0  = (blk % (W_ / TW)) * TW;
    const int tid = threadIdx.x;

    const float4* __restrict__ subB = sub   + (size_t)b * (W_ * QPR);
    const float4* __restrict__ sdB  = subed + (size_t)b * (W_ * QPR);
    float4*       __restrict__ outB = out   + (size_t)b * (W_ * QPR);

    // ---- Stage sub rows [w0-8, w0+TW+8) into LDS (clamped at edges) --------
    // Each iteration is one float4 (16B) per lane, fully coalesced.
    for (int i = tid; i < ROWS * QPR; i += NTHREADS) {
        const int row  = i >> 4;        // / QPR
        const int q    = i & (QPR - 1);
        int grow = w0 - HALO + row;
        grow = grow < 0 ? 0 : (grow > W_ - 1 ? W_ - 1 : grow);
        const float4* gp = subB + grow * QPR + q;
#if HAVE_ASYNC_LDS
        __builtin_amdgcn_global_load_async_to_lds_b128(
            (v4i*)gp, (v4i*)&lds[i], 0, 0);
#else
        lds[i] = *gp;
#endif
    }
#if HAVE_ASYNC_LDS
    wait_async_zero();
#endif
    __syncthreads();

    // ---- Compute: each thread owns (quad q, one row per pass) --------------
    const int q  = tid & (QPR - 1);
    const int r0 = tid >> 4;            // 0..15, 16 rows per pass

#pragma unroll
    for (int pass = 0; pass < TW / 16; ++pass) {
        const int lr = r0 + pass * 16;  // local row 0..TW-1
        const int w  = w0 + lr;

        // Prefetch the streaming subed read for the next pass (gfx1250
        // global_prefetch_b8); harmless on the last pass (clamped).
        {
            int wp = w + 16; wp = wp > W_ - 1 ? W_ - 1 : wp;
            __builtin_prefetch((const void*)(sdB + (size_t)wp * QPR + q), 0, 0);
        }

        const float4 sv = sdB[(size_t)w * QPR + q];
        const float4* base = &lds[(lr + HALO) * QPR + q];

        // delay 0 (j = 0)
        float4 bv, ba;
        {
            const float4 g = base[0];
            bv.x = sv.x - g.x; ba.x = __builtin_fabsf(bv.x);
            bv.y = sv.y - g.y; ba.y = __builtin_fabsf(bv.y);
            bv.z = sv.z - g.z; ba.z = __builtin_fabsf(bv.z);
            bv.w = sv.w - g.w; ba.w = __builtin_fabsf(bv.w);
        }
        // delays +1..+8 (j = 1..8)
#pragma unroll
        for (int dd = 1; dd <= D_; ++dd) {
            const float4 g = base[dd * QPR];
            UPD_ALL();
        }
        // delays -1..-8 (j = 9..16)
#pragma unroll
        for (int dd = 1; dd <= D_; ++dd) {
            const float4 g = base[-dd * QPR];
            UPD_ALL();
        }

        outB[(size_t)w * QPR + q] = bv;
    }
}

extern "C" void kernel_launch(void* const* d_in, const int* in_sizes, int n_in,
                              void* d_out, int out_size, void* d_ws, size_t ws_size,
                              hipStream_t stream) {
    (void)in_sizes; (void)n_in; (void)d_ws; (void)ws_size; (void)out_size;
    const float4* subed = (const float4*)d_in[0];
    const float4* sub   = (const float4*)d_in[1];
    // d_in[2] is the scalar delay count d == 8; baked in as D_ above.
    float4* out = (float4*)d_out;

    dim3 grid(B_ * (W_ / TW));   // 8 * 64 = 512 blocks
    dim3 block(NTHREADS);
    OffsetSubtraction_47785806135946_kernel<<<grid, block, 0, stream>>>(subed, sub, out);
}